// BaseGinENetwork_83073257439839
// MI455X (gfx1250) — compile-verified
//
#include <hip/hip_runtime.h>

// ---------------- problem constants ----------------
#define N_JOBS  2048
#define N_MACH  256
#define HID     128
#define EMB     64
#define N_JM    (N_JOBS * N_MACH)      // 524288
#define N_JJ    (N_JOBS - 1)           // 2047
#define N_E     (N_JM + N_JJ)          // 526335
#define N_NODES (N_JOBS + N_MACH)      // 2304
#define BN_EPS  1e-5f

typedef __attribute__((ext_vector_type(2))) float v2f;
typedef __attribute__((ext_vector_type(8))) float v8f;

// ---------------- tiny utility kernels ----------------
__global__ void k_fill_f32(float* p, float v, long n) {
  long i = (long)blockIdx.x * blockDim.x + threadIdx.x;
  if (i < n) p[i] = v;
}
__global__ void k_fill_i32(int* p, int v, long n) {
  long i = (long)blockIdx.x * blockDim.x + threadIdx.x;
  if (i < n) p[i] = v;
}
__global__ void k_copy_f32(const float* a, float* o, long n) {
  long i = (long)blockIdx.x * blockDim.x + threadIdx.x;
  if (i < n) o[i] = a[i];
}
__global__ void k_add_f32(const float* a, const float* b, float* o, long n) {
  long i = (long)blockIdx.x * blockDim.x + threadIdx.x;
  if (i < n) o[i] = a[i] + b[i];
}

// ---------------- ordered nonzero compaction (jnp.nonzero semantics) ----------
__global__ void k_nz_count(const int* flags, long n, int* blkcnt) {
  __shared__ int s;
  if (threadIdx.x == 0) s = 0;
  __syncthreads();
  long i = (long)blockIdx.x * 1024 + threadIdx.x;
  if (i < n && flags[i] != 0) atomicAdd(&s, 1);
  __syncthreads();
  if (threadIdx.x == 0) blkcnt[blockIdx.x] = s;
}
__global__ void k_nz_exscan(int* cnt, int nblocks) {
  if (threadIdx.x == 0 && blockIdx.x == 0) {
    int acc = 0;
    for (int b = 0; b < nblocks; ++b) { int c = cnt[b]; cnt[b] = acc; acc += c; }
  }
}
__global__ void k_nz_compact(const int* flags, long n, const int* offs,
                             int* outIdx, int maxOut) {
  __shared__ int sdata[1024];
  int tid = threadIdx.x;
  long i = (long)blockIdx.x * 1024 + tid;
  int f = (i < n && flags[i] != 0) ? 1 : 0;
  sdata[tid] = f;
  __syncthreads();
  for (int d = 1; d < 1024; d <<= 1) {
    int v = (tid >= d) ? sdata[tid - d] : 0;
    __syncthreads();
    sdata[tid] += v;
    __syncthreads();
  }
  if (f) {
    int pos = offs[blockIdx.x] + sdata[tid] - 1;
    if (pos < maxOut) outIdx[pos] = (int)i;
  }
}

// ---------------- edge table + features ----------------
__global__ void k_build_edges(const int* __restrict__ jm_idx, const int* __restrict__ jj_pos,
                              const float* __restrict__ tvt, const float* __restrict__ tve,
                              int* __restrict__ src, int* __restrict__ dst,
                              float* __restrict__ ef) {
  long i = (long)blockIdx.x * blockDim.x + threadIdx.x;
  if (i < N_JM) {
    int idx = jm_idx[i];
    src[i] = idx / N_MACH;
    dst[i] = idx % N_MACH + N_JOBS;
    ef[i * 3 + 0] = 1.0f;
    ef[i * 3 + 1] = tvt[idx];
    ef[i * 3 + 2] = tve[idx];
  } else if (i < N_E) {
    int p = jj_pos[i - N_JM];
    src[i] = p / N_JOBS;
    dst[i] = p % N_JOBS;
    ef[i * 3 + 0] = 0.0f;
    ef[i * 3 + 1] = 0.0f;
    ef[i * 3 + 2] = 0.0f;
  }
}

__global__ void k_stack3(const float* a, const float* b, const float* c, float* o, int n) {
  int i = blockIdx.x * blockDim.x + threadIdx.x;
  if (i < n) { o[i * 3] = a[i]; o[i * 3 + 1] = b[i]; o[i * 3 + 2] = c[i]; }
}

// ---------------- small-K GEMM (K = 1 or 3), W staged in LDS ----------------
__global__ void k_gemm_smallk(const float* __restrict__ A, const float* __restrict__ W,
                              const float* __restrict__ bias, float* __restrict__ C,
                              long M, int K, int N) {
  __shared__ float sW[512];
  __shared__ float sB[128];
  for (int i = threadIdx.x; i < K * N; i += blockDim.x) sW[i] = W[i];
  for (int i = threadIdx.x; i < N; i += blockDim.x) sB[i] = bias[i];
  __syncthreads();
  long i = (long)blockIdx.x * blockDim.x + threadIdx.x;
  if (i >= M * (long)N) return;
  long r = i / N;
  int  c = (int)(i % N);
  float acc = sB[c];
  for (int k = 0; k < K; ++k) acc += A[r * K + k] * sW[k * N + c];
  C[i] = acc;
}

// ---------------- BatchNorm column stats (sum / sumsq) ----------------
__global__ void k_bn_stats(const float* __restrict__ X, long M, int N,
                           float* __restrict__ sum, float* __restrict__ sumsq,
                           int rowsPerBlock) {
  int c = threadIdx.x;                       // blockDim.x == N
  long r0 = (long)blockIdx.x * rowsPerBlock;
  long r1 = r0 + rowsPerBlock; if (r1 > M) r1 = M;
  float s = 0.f, s2 = 0.f;
  for (long r = r0; r < r1; ++r) {
    if (r + 8 < r1) __builtin_prefetch(&X[(r + 8) * N + c], 0, 0);
    float v = X[r * N + c];
    s += v; s2 += v * v;
  }
  unsafeAtomicAdd(&sum[c], s);
  unsafeAtomicAdd(&sumsq[c], s2);
}

// ---------------- WMMA f32 GEMM: C = act((BNA? bnrelu(A) : A) @ W + b) -------
// W held in LDS pair-interleaved: element (k,n) -> sW[(k>>1)*S + 2n + (k&1)],
// S = 2N+32 floats, so each B fragment is one aligned ds_load_b64 and the two
// half-waves (pair-rows p and p+1) land on disjoint 32-bank groups.
template <int N, bool BNA>
__global__ void k_wmma_gemm_t(const float* __restrict__ A, const float* __restrict__ W,
                              const float* __restrict__ bias,
                              const float* __restrict__ bnSum, const float* __restrict__ bnSq,
                              const float* __restrict__ bnG, const float* __restrict__ bnBe,
                              float* __restrict__ C, long M, int K, int relu) {
  constexpr int NT = N / 16;
  constexpr int S  = 2 * N + 32;
  extern __shared__ float smem[];
  float* sW     = smem;                      // (K/2)*S
  float* sB     = sW + (size_t)(K / 2) * S;  // N
  float* sScale = sB + N;                    // K
  float* sShift = sScale + K;                // K
  for (int i = threadIdx.x; i < K * N; i += 256) {
    int k = i / N, n = i % N;
    sW[(k >> 1) * S + 2 * n + (k & 1)] = W[i];
  }
  for (int i = threadIdx.x; i < N; i += 256) sB[i] = bias[i];
  if (BNA) {
    for (int k = threadIdx.x; k < K; k += 256) {
      float mean = bnSum[k] / (float)M;
      float var  = bnSq[k] / (float)M - mean * mean;
      float sc   = bnG[k] * rsqrtf(var + BN_EPS);
      sScale[k] = sc;
      sShift[k] = bnBe[k] - mean * sc;
    }
  }
  __syncthreads();

  int wave = threadIdx.x >> 5;
  int lane = threadIdx.x & 31;
  long row0 = ((long)blockIdx.x * 8 + wave) * 16;
  if (row0 >= M) return;                    // wave-uniform exit
  int mrow = lane & 15;
  int kb   = (lane >> 4) * 2;               // 0 or 2
  long arow = row0 + mrow; if (arow > M - 1) arow = M - 1;
  const float* Arow = A + arow * K;

  v8f acc[NT];
#pragma unroll
  for (int t = 0; t < NT; ++t) acc[t] = (v8f){0.f,0.f,0.f,0.f,0.f,0.f,0.f,0.f};

  for (int k0 = 0; k0 < K; k0 += 4) {
    float ax = Arow[k0 + kb];
    float ay = Arow[k0 + kb + 1];
    if (BNA) {
      ax = fmaxf(fmaf(ax, sScale[k0 + kb],     sShift[k0 + kb]),     0.f);
      ay = fmaxf(fmaf(ay, sScale[k0 + kb + 1], sShift[k0 + kb + 1]), 0.f);
    }
    v2f a; a.x = ax; a.y = ay;
    const float* bprow = sW + ((k0 + kb) >> 1) * S + 2 * mrow;
#pragma unroll
    for (int t = 0; t < NT; ++t) {
      v2f b = *(const v2f*)(bprow + 32 * t);    // (k0+kb, col), (k0+kb+1, col)
      acc[t] = __builtin_amdgcn_wmma_f32_16x16x4_f32(false, a, false, b, (short)0,
                                                     acc[t], false, false);
    }
  }
#pragma unroll
  for (int t = 0; t < NT; ++t) {
    int col = t * 16 + mrow;
    float bv = sB[col];
#pragma unroll
    for (int v = 0; v < 8; ++v) {
      long r = row0 + v + ((lane >> 4) << 3);
      if (r < M) {
        float val = acc[t][v] + bv;
        if (relu) val = fmaxf(val, 0.f);
        C[r * N + col] = val;
      }
    }
  }
}

template <int N>
static void launch_gemm(const float* A, const float* W, const float* bias,
                        const float* bs, const float* bss, const float* g, const float* be,
                        float* C, long M, int K, int relu, hipStream_t stream) {
  long rowTiles = (M + 15) / 16;
  unsigned blocks = (unsigned)((rowTiles + 7) / 8);
  size_t sh = ((size_t)(K / 2) * (2 * N + 32) + N + 2 * (size_t)K) * sizeof(float);
  if (bs)
    k_wmma_gemm_t<N, true><<<blocks, 256, sh, stream>>>(A, W, bias, bs, bss, g, be, C, M, K, relu);
  else
    k_wmma_gemm_t<N, false><<<blocks, 256, sh, stream>>>(A, W, bias, bs, bss, g, be, C, M, K, relu);
}

// ---------------- fused GIN edge phase (templated on din):
//   msg = relu(x[src] + e @ We + eb);  agg[dst] += msg  (global_atomic_add_f32)
template <int DIN>
__global__ void k_edge_msg_t(const float* __restrict__ E, const float* __restrict__ We,
                             const float* __restrict__ eb, const float* __restrict__ X,
                             const int* __restrict__ src, const int* __restrict__ dst,
                             float* __restrict__ agg) {
  constexpr int NT = DIN / 16;
  constexpr int S  = 2 * DIN + 32;
  extern __shared__ float smem[];
  float* sWe = smem;                     // (EMB/2)*S, pair-interleaved
  float* sEb = sWe + (EMB / 2) * S;      // DIN
  for (int i = threadIdx.x; i < EMB * DIN; i += 256) {
    int k = i / DIN, n = i % DIN;
    sWe[(k >> 1) * S + 2 * n + (k & 1)] = We[i];
  }
  for (int i = threadIdx.x; i < DIN; i += 256) sEb[i] = eb[i];
  __syncthreads();

  int wave = threadIdx.x >> 5;
  int lane = threadIdx.x & 31;
  long e0 = ((long)blockIdx.x * 8 + wave) * 16;   // 16-edge group per wave
  if (e0 >= N_E) return;                          // wave-uniform exit
  int mrow = lane & 15;
  int kb   = (lane >> 4) * 2;
  long erow = e0 + mrow; if (erow > (long)N_E - 1) erow = N_E - 1;
  const float* Erow = E + erow * EMB;

  v8f acc[NT];
#pragma unroll
  for (int t = 0; t < NT; ++t) acc[t] = (v8f){0.f,0.f,0.f,0.f,0.f,0.f,0.f,0.f};

#pragma unroll 4
  for (int k0 = 0; k0 < EMB; k0 += 4) {
    v2f a;
    a.x = Erow[k0 + kb];
    a.y = Erow[k0 + kb + 1];
    const float* bprow = sWe + ((k0 + kb) >> 1) * S + 2 * mrow;
#pragma unroll
    for (int t = 0; t < NT; ++t) {
      v2f b = *(const v2f*)(bprow + 32 * t);
      acc[t] = __builtin_amdgcn_wmma_f32_16x16x4_f32(false, a, false, b, (short)0,
                                                     acc[t], false, false);
    }
  }
#pragma unroll
  for (int v = 0; v < 8; ++v) {
    long er = e0 + v + ((lane >> 4) << 3);
    if (er < N_E) {
      int s = src[er], d = dst[er];
      const float* xr = X + (long)s * DIN;
      float* ar = agg + (long)d * DIN;
#pragma unroll
      for (int t = 0; t < NT; ++t) {
        int col = t * 16 + mrow;
        float val = acc[t][v] + sEb[col] + xr[col];
        unsafeAtomicAdd(&ar[col], fmaxf(val, 0.f));
      }
    }
  }
}

template <int DIN>
static void launch_edge_msg(const float* E, const float* We, const float* eb,
                            const float* X, const int* src, const int* dst,
                            float* agg, hipStream_t stream) {
  long tiles = ((long)N_E + 15) / 16;
  unsigned blocks = (unsigned)((tiles + 7) / 8);
  size_t sh = ((size_t)(EMB / 2) * (2 * DIN + 32) + DIN) * sizeof(float);
  k_edge_msg_t<DIN><<<blocks, 256, sh, stream>>>(E, We, eb, X, src, dst, agg);
}

// ---------------- outputs ----------------
__global__ void k_edge_emb(const float* __restrict__ x, const int* __restrict__ src,
                           const int* __restrict__ dst, float* __restrict__ out) {
  long i = (long)blockIdx.x * blockDim.x + threadIdx.x;
  if (i >= (long)N_E * 128) return;
  long e = i >> 7;
  int  c = (int)(i & 127);
  int node = (c < EMB) ? src[e] : dst[e];
  out[i] = x[(long)node * EMB + (c & 63)];
}
__global__ void k_graph_mean(const float* __restrict__ x, float* __restrict__ o) {
  int c = threadIdx.x;                      // 64 threads
  float s = 0.f;
  for (int r = 0; r < N_NODES; ++r) s += x[(long)r * EMB + c];
  o[c] = s / (float)N_NODES;
}

// ---------------- host orchestration ----------------
static inline long cdiv(long a, long b) { return (a + b - 1) / b; }

extern "C" void kernel_launch(void* const* d_in, const int* in_sizes, int n_in,
                              void* d_out, int out_size, void* d_ws, size_t ws_size,
                              hipStream_t stream) {
  (void)in_sizes; (void)n_in; (void)out_size; (void)ws_size;
  const float* tss = (const float*)d_in[0];
  const float* tsr = (const float*)d_in[1];
  const float* tct = (const float*)d_in[2];
  const float* vct = (const float*)d_in[3];
  const float* tvt = (const float*)d_in[4];
  const float* tve = (const float*)d_in[5];
  const int*   compat = (const int*)d_in[6];
  const int*   adj    = (const int*)d_in[7];
  const float** P = (const float**)&d_in[8];  // params in dict insertion order
  // job: W1 b1 g1 be1 W2 b2 g2 be2 W3 b3     -> P[0..9]
  // mach: W1 b1 g1 be1 W2 b2                 -> P[10..15]
  // edge: W1 b1 g1 be1 W2 b2                 -> P[16..21]
  // gin i: We eb W1 b1 g1 beta1 W2 b2        -> P[22+8*(i-1) ...]
  float* out = (float*)d_out;

  char* w = (char*)d_ws;
  auto alloc = [&](size_t bytes) -> void* {
    void* p = (void*)w;
    w += (bytes + 255) & ~(size_t)255;
    return p;
  };
  int*   src    = (int*)alloc((size_t)N_E * 4);
  int*   dst    = (int*)alloc((size_t)N_E * 4);
  int*   jm_idx = (int*)alloc((size_t)N_JM * 4);
  int*   jj_pos = (int*)alloc((size_t)N_JJ * 4);
  int*   blkcnt = (int*)alloc(4096 * 4);
  float* ef     = (float*)alloc((size_t)N_E * 3 * 4);
  float* he     = (float*)alloc((size_t)N_E * HID * 4);   // 269 MB
  float* eemb   = (float*)alloc((size_t)N_E * EMB * 4);   // 135 MB
  float* xbuf   = (float*)alloc((size_t)N_NODES * HID * 4);
  float* hbuf   = (float*)alloc((size_t)N_NODES * HID * 4);
  float* tbuf   = (float*)alloc((size_t)N_NODES * HID * 4);
  float* jobf   = (float*)alloc((size_t)N_JOBS * 3 * 4);
  float* sums   = (float*)alloc(HID * 4);
  float* sumsq  = (float*)alloc(HID * 4);

  auto bnstats = [&](const float* X, long M, int N) {
    k_fill_f32<<<1, 256, 0, stream>>>(sums, 0.f, HID);
    k_fill_f32<<<1, 256, 0, stream>>>(sumsq, 0.f, HID);
    int rpb = (M > 4096) ? 512 : 64;
    k_bn_stats<<<dim3((unsigned)cdiv(M, rpb)), N, 0, stream>>>(X, M, N, sums, sumsq, rpb);
  };

  // ---- 1. nonzero(compat) -> jm_idx (row-major order, zero-padded) ----
  k_fill_i32<<<dim3((unsigned)cdiv(N_JM, 256)), 256, 0, stream>>>(jm_idx, 0, N_JM);
  k_nz_count<<<512, 1024, 0, stream>>>(compat, N_JM, blkcnt);
  k_nz_exscan<<<1, 1, 0, stream>>>(blkcnt, 512);
  k_nz_compact<<<512, 1024, 0, stream>>>(compat, N_JM, blkcnt, jm_idx, N_JM);

  // ---- 2. nonzero(adj) -> jj positions ----
  k_fill_i32<<<dim3((unsigned)cdiv(N_JJ, 256)), 256, 0, stream>>>(jj_pos, 0, N_JJ);
  k_nz_count<<<4096, 1024, 0, stream>>>(adj, (long)N_JOBS * N_JOBS, blkcnt);
  k_nz_exscan<<<1, 1, 0, stream>>>(blkcnt, 4096);
  k_nz_compact<<<4096, 1024, 0, stream>>>(adj, (long)N_JOBS * N_JOBS, blkcnt, jj_pos, N_JJ);

  // ---- 3. edge table + edge features ----
  k_build_edges<<<dim3((unsigned)cdiv(N_E, 256)), 256, 0, stream>>>(jm_idx, jj_pos, tvt, tve, src, dst, ef);

  // ---- 4. edge encoder: e = relu(bn(ef@W1+b1)) @ W2 + b2  (BN fused into GEMM) ----
  k_gemm_smallk<<<dim3((unsigned)cdiv((long)N_E * HID, 256)), 256, 0, stream>>>(ef, P[16], P[17], he, N_E, 3, HID);
  bnstats(he, N_E, HID);
  launch_gemm<EMB>(he, P[20], P[21], sums, sumsq, P[18], P[19], eemb, N_E, HID, 0, stream);

  // ---- 5. job encoder -> x rows [0, 2048) ----
  k_stack3<<<dim3((unsigned)cdiv(N_JOBS, 256)), 256, 0, stream>>>(tss, tsr, tct, jobf, N_JOBS);
  k_gemm_smallk<<<dim3((unsigned)cdiv((long)N_JOBS * HID, 256)), 256, 0, stream>>>(jobf, P[0], P[1], hbuf, N_JOBS, 3, HID);
  bnstats(hbuf, N_JOBS, HID);
  launch_gemm<HID>(hbuf, P[4], P[5], sums, sumsq, P[2], P[3], tbuf, N_JOBS, HID, 0, stream);
  bnstats(tbuf, N_JOBS, HID);
  launch_gemm<EMB>(tbuf, P[8], P[9], sums, sumsq, P[6], P[7], xbuf, N_JOBS, HID, 0, stream);

  // ---- 6. machine encoder -> x rows [2048, 2304) ----
  k_gemm_smallk<<<dim3((unsigned)cdiv((long)N_MACH * HID, 256)), 256, 0, stream>>>(vct, P[10], P[11], hbuf, N_MACH, 1, HID);
  bnstats(hbuf, N_MACH, HID);
  launch_gemm<EMB>(hbuf, P[14], P[15], sums, sumsq, P[12], P[13],
                   xbuf + (long)N_JOBS * EMB, N_MACH, HID, 0, stream);

  // ---- 7. three GIN layers ----
  const int dins[3]  = {EMB, HID, HID};
  const int douts[3] = {HID, HID, EMB};
  for (int l = 0; l < 3; ++l) {
    const float** G = &P[22 + 8 * l];  // We eb W1 b1 g1 beta1 W2 b2
    int din = dins[l], dout = douts[l];
    float* agg = tbuf;
    k_fill_f32<<<dim3((unsigned)cdiv((long)N_NODES * din, 256)), 256, 0, stream>>>(agg, 0.f, (long)N_NODES * din);
    if (din == EMB)
      launch_edge_msg<EMB>(eemb, G[0], G[1], xbuf, src, dst, agg, stream);
    else
      launch_edge_msg<HID>(eemb, G[0], G[1], xbuf, src, dst, agg, stream);
    k_add_f32<<<dim3((unsigned)cdiv((long)N_NODES * din, 256)), 256, 0, stream>>>(xbuf, agg, hbuf, (long)N_NODES * din);
    // h1 = h @ W1 + b1 (raw; BN fused into next GEMM)
    if (dout == HID)
      launch_gemm<HID>(hbuf, G[2], G[3], nullptr, nullptr, nullptr, nullptr, tbuf, N_NODES, din, 0, stream);
    else
      launch_gemm<EMB>(hbuf, G[2], G[3], nullptr, nullptr, nullptr, nullptr, tbuf, N_NODES, din, 0, stream);
    bnstats(tbuf, N_NODES, dout);
    // x = act(relu(bn(h1)) @ W2 + b2)
    if (dout == HID)
      launch_gemm<HID>(tbuf, G[6], G[7], sums, sumsq, G[4], G[5], xbuf, N_NODES, dout, (l < 2) ? 1 : 0, stream);
    else
      launch_gemm<EMB>(tbuf, G[6], G[7], sums, sumsq, G[4], G[5], xbuf, N_NODES, dout, (l < 2) ? 1 : 0, stream);
  }

  // ---- 8. outputs: node_emb | edge_emb | graph_emb ----
  k_copy_f32<<<dim3((unsigned)cdiv((long)N_NODES * EMB, 256)), 256, 0, stream>>>(xbuf, out, (long)N_NODES * EMB);
  k_edge_emb<<<dim3((unsigned)cdiv((long)N_E * 128, 256)), 256, 0, stream>>>(xbuf, src, dst, out + (long)N_NODES * EMB);
  k_graph_mean<<<1, EMB, 0, stream>>>(xbuf, out + (long)N_NODES * EMB + (long)N_E * 128);
}